// SimplifiedSSM_22076131902202
// MI455X (gfx1250) — compile-verified
//
#include <hip/hip_runtime.h>
#include <math.h>

typedef __attribute__((ext_vector_type(2))) float v2f;
typedef __attribute__((ext_vector_type(8))) float v8f;

// Low 32 bits of a generic pointer into LDS == LDS byte offset (ISA aperture rule).
__device__ __forceinline__ uint32_t lds_off(const void* p) {
  return (uint32_t)(uintptr_t)p;
}

// CDNA5 async global->LDS 16B copy (ASYNCcnt-tracked, op 98).
__device__ __forceinline__ void async_copy_b128(uint32_t lds_byte_off,
                                                const void* gsrc) {
  asm volatile("global_load_async_to_lds_b128 %0, %1, off"
               :
               : "v"(lds_byte_off), "v"((uint64_t)(uintptr_t)gsrc)
               : "memory");
}

__device__ __forceinline__ void wait_async0() {
  asm volatile("s_wait_asynccnt 0" ::: "memory");
}

// ---------------------------------------------------------------------------
// FP32 WMMA GEMM: out = A[M,K] @ W[K,N] + bias, tiled 128 x (16*NSUB).
// 8 waves per block stacked in M; each wave owns a 16-row strip and NSUB
// 16x16 f32 accumulators (v8f each), fed by V_WMMA_F32_16X16X4_F32.
// A tile staged via async global->LDS copies; B tile stored K-pair-interleaved
// so every B fragment is a single aligned ds_load_b64. All fragments for a
// BK chunk are prefetched to registers, then 4*NSUB WMMAs issue back-to-back.
// MODE 0: out0[m,n] = acc + bias[n]
// MODE 1: n <  splitN : out0[m,n]        = silu(acc + bias[n])
//         n >= splitN : out1[m,n-splitN] =      acc + bias[n]
// ---------------------------------------------------------------------------
template <int NSUB, int MODE>
__global__ __launch_bounds__(256) void gemm_wmma_f32(
    const float* __restrict__ A, const float* __restrict__ W,
    const float* __restrict__ bias, float* __restrict__ out0,
    float* __restrict__ out1, int M, int N, int K, int splitN) {
  constexpr int BM = 128;
  constexpr int BK = 16;
  constexpr int BN = 16 * NSUB;
  constexpr int AS = BK + 4;  // 20 floats: 80B row pitch -> 16B aligned, conflict-free

  __shared__ float ldsA[BM * AS];
  // B tile in fragment order: [BK/2][BN] float2 = { W[k0+2kp][n], W[k0+2kp+1][n] }
  __shared__ float ldsBp[BK * BN];

  const int tid  = threadIdx.x;
  const int wave = tid >> 5;
  const int lane = tid & 31;
  const int half = lane >> 4;  // K-pair select (A/B), M-half (C/D)
  const int nloc = lane & 15;

  const int tileM = blockIdx.x * BM;
  const int tileN = blockIdx.y * BN;

  const v8f vzero = {0.f, 0.f, 0.f, 0.f, 0.f, 0.f, 0.f, 0.f};
  v8f acc[NSUB];
#pragma unroll
  for (int j = 0; j < NSUB; ++j) acc[j] = vzero;

  const int arow = wave * 16 + nloc;  // this lane's A row within the block tile
  const uint32_t ldsA_base = lds_off(&ldsA[0]);

  for (int k0 = 0; k0 < K; k0 += BK) {
    __syncthreads();  // previous iteration's readers done before overwriting LDS

    // Stage A tile [BM x BK]: exactly 2 contiguous 16B async copies per thread.
#pragma unroll
    for (int it = 0; it < (BM * BK) / 4 / 256; ++it) {
      const int i   = tid + it * 256;
      const int row = i >> 2;          // BK/4 == 4
      const int c4  = (i & 3) * 4;
      async_copy_b128(ldsA_base + (uint32_t)((row * AS + c4) * 4),
                      &A[(size_t)(tileM + row) * K + k0 + c4]);
    }
    // Stage B tile [BK x BN] into K-pair-interleaved fragment layout.
    constexpr int BQ = (BK * BN) / 4;  // float4 transfer count
    {
      const int i = tid;
      const bool active = (BQ >= 256) || (tid < BQ);
      if (active) {
        const int krow = i / (BN / 4);
        const int c4   = (i % (BN / 4)) * 4;
        const float4 v = *(const float4*)&W[(size_t)(k0 + krow) * N + tileN + c4];
        const int kp = krow >> 1, klo = krow & 1;
        ldsBp[(kp * BN + c4 + 0) * 2 + klo] = v.x;
        ldsBp[(kp * BN + c4 + 1) * 2 + klo] = v.y;
        ldsBp[(kp * BN + c4 + 2) * 2 + klo] = v.z;
        ldsBp[(kp * BN + c4 + 3) * 2 + klo] = v.w;
      }
    }
    wait_async0();   // A-tile async copies landed in LDS
    __syncthreads(); // all staging visible block-wide

    // Prefetch all fragments for this BK chunk, then issue WMMAs back-to-back.
    v2f afrag[4];
    v2f bfrag[4][NSUB];
#pragma unroll
    for (int kk = 0; kk < 4; ++kk) {
      const int kb2 = kk * 2 + half;  // this lane's K-pair index
      afrag[kk] = *(const v2f*)&ldsA[arow * AS + kb2 * 2];
#pragma unroll
      for (int j = 0; j < NSUB; ++j)
        bfrag[kk][j] = *(const v2f*)&ldsBp[(kb2 * BN + j * 16 + nloc) * 2];
    }
#pragma unroll
    for (int kk = 0; kk < 4; ++kk) {
#pragma unroll
      for (int j = 0; j < NSUB; ++j) {
        acc[j] = __builtin_amdgcn_wmma_f32_16x16x4_f32(
            false, afrag[kk], false, bfrag[kk][j], (short)0, acc[j], false,
            false);
      }
    }
  }

  // Epilogue: C/D layout -> element (m = r + 8*half, n = nloc) in acc[j][r]
#pragma unroll
  for (int j = 0; j < NSUB; ++j) {
    const int col = tileN + j * 16 + nloc;
    const float bs = bias[col];
#pragma unroll
    for (int r = 0; r < 8; ++r) {
      const int row = tileM + wave * 16 + r + 8 * half;
      const float v = acc[j][r] + bs;
      if (MODE == 1) {
        if (col < splitN) {
          out0[(size_t)row * splitN + col] = v / (1.0f + __expf(-v));  // silu
        } else {
          out1[(size_t)row * splitN + (col - splitN)] = v;
        }
      } else {
        out0[(size_t)row * N + col] = v;
      }
    }
  }
}

// ---------------------------------------------------------------------------
// Fused conv1d(3, depthwise, pad=1) + selective scan + D-skip + gate multiply.
// One thread per (b, d) chain; 16 states in registers; BC[t,0:32] staged per
// 64-step chunk into LDS via async copies (uniform broadcast reads in-wave).
// ---------------------------------------------------------------------------
__global__ __launch_bounds__(64) void scan_fused(
    const float* __restrict__ x_input, const float* __restrict__ x_gate,
    const float* __restrict__ BC, const float* __restrict__ A_log,
    const float* __restrict__ D_param, const float* __restrict__ conv_w,
    const float* __restrict__ conv_b, float* __restrict__ y, int L, int D) {
  constexpr int TB = 64;  // timesteps per LDS chunk
  constexpr int NS = 16;
  __shared__ float ldsBC[TB * 2 * NS];  // [TB][32]

  const int b = blockIdx.x;
  const int d = blockIdx.y * 64 + threadIdx.x;

  float Aa[NS], s[NS];
#pragma unroll
  for (int n = 0; n < NS; ++n) {
    Aa[n] = -__expf(A_log[n]);
    s[n] = 0.f;
  }
  const float w0 = conv_w[d * 3 + 0];
  const float w1 = conv_w[d * 3 + 1];
  const float w2 = conv_w[d * 3 + 2];
  const float cb = conv_b[d];
  const float Dp = D_param[d];

  const size_t base = ((size_t)b * L) * D + d;
  float xm1 = 0.f;                 // x_input[t-1]
  float x0  = x_input[base];       // x_input[t]
  const uint32_t ldsBC_base = lds_off(&ldsBC[0]);

  for (int tc = 0; tc < L; tc += TB) {
    __syncthreads();
    const float* src = BC + ((size_t)b * L + tc) * (2 * NS);
#pragma unroll
    for (int it = 0; it < (TB * 2 * NS) / 4 / 64; ++it) {  // exactly 8 copies
      const int i = threadIdx.x + it * 64;
      async_copy_b128(ldsBC_base + (uint32_t)(i * 16), &src[i * 4]);
    }
    wait_async0();
    __syncthreads();

#pragma unroll 4
    for (int tt = 0; tt < TB; ++tt) {
      const int t = tc + tt;
      const float xp1 = (t + 1 < L) ? x_input[base + (size_t)(t + 1) * D] : 0.f;
      const float u = fmaf(w0, xm1, fmaf(w1, x0, fmaf(w2, xp1, cb)));
      const float* bc = &ldsBC[tt * 2 * NS];
      float accum = 0.f;
#pragma unroll
      for (int n = 0; n < NS; ++n) {
        s[n] = fmaf(Aa[n], s[n], bc[n] * u);        // state update
        accum = fmaf(bc[NS + n], s[n], accum);      // y_t = C_t . s_t
      }
      const float g = x_gate[base + (size_t)t * D];
      y[base + (size_t)t * D] = fmaf(Dp, x0, accum) * g;
      xm1 = x0;
      x0 = xp1;
    }
  }
}

// ---------------------------------------------------------------------------
extern "C" void kernel_launch(void* const* d_in, const int* in_sizes, int n_in,
                              void* d_out, int out_size, void* d_ws,
                              size_t ws_size, hipStream_t stream) {
  (void)in_sizes; (void)n_in; (void)out_size; (void)ws_size;

  constexpr int Bb = 8, L = 2048, D = 1024, NS = 16;
  constexpr int M = Bb * L;  // 16384

  const float* x       = (const float*)d_in[0];
  const float* W_in    = (const float*)d_in[1];
  const float* b_in    = (const float*)d_in[2];
  const float* W_x     = (const float*)d_in[3];
  const float* b_x     = (const float*)d_in[4];
  const float* W_out   = (const float*)d_in[5];
  const float* b_out   = (const float*)d_in[6];
  const float* A_log   = (const float*)d_in[7];
  const float* D_param = (const float*)d_in[8];
  const float* conv_w  = (const float*)d_in[9];
  const float* conv_b  = (const float*)d_in[10];
  float* out = (float*)d_out;

  float* ws      = (float*)d_ws;
  float* x_gate  = ws;                            // M*D
  float* x_input = x_gate  + (size_t)M * D;       // M*D
  float* y_mid   = x_input + (size_t)M * D;       // M*D
  float* BCbuf   = y_mid   + (size_t)M * D;       // M*2*NS

  // 1) x_proj = x @ W_in + b_in  ->  silu gate + x_input halves
  gemm_wmma_f32<4, 1><<<dim3(M / 128, (2 * D) / 64), 256, 0, stream>>>(
      x, W_in, b_in, x_gate, x_input, M, 2 * D, D, D);

  // 2) BC = x_input @ W_x + b_x   (N = 32)
  gemm_wmma_f32<2, 0><<<dim3(M / 128, (2 * NS) / 32), 256, 0, stream>>>(
      x_input, W_x, b_x, BCbuf, nullptr, M, 2 * NS, D, 0);

  // 3) fused conv + scan + skip + gate  ->  y_mid
  scan_fused<<<dim3(Bb, D / 64), 64, 0, stream>>>(
      x_input, x_gate, BCbuf, A_log, D_param, conv_w, conv_b, y_mid, L, D);

  // 4) out = y_mid @ W_out + b_out
  gemm_wmma_f32<4, 0><<<dim3(M / 128, D / 64), 256, 0, stream>>>(
      y_mid, W_out, b_out, out, nullptr, M, D, D, 0);
}